// Router_46935402611125
// MI455X (gfx1250) — compile-verified
//
#include <hip/hip_runtime.h>
#include <math.h>

typedef __attribute__((ext_vector_type(2))) float v2f;
typedef __attribute__((ext_vector_type(4))) float v4f;
typedef __attribute__((ext_vector_type(8))) float v8f;

#define N_EXP 8
#define TOP_K 2

// ---------------- Phase 0: streaming zero-fill of the 536MB output ----------
__global__ void router_zero_kernel(float* __restrict__ out, size_t n) {
    const size_t n4 = n >> 2;
    v4f z = {0.f, 0.f, 0.f, 0.f};
    v4f* p4 = (v4f*)out;
    for (size_t i = blockIdx.x * (size_t)blockDim.x + threadIdx.x; i < n4;
         i += (size_t)gridDim.x * blockDim.x) {
        __builtin_nontemporal_store(z, p4 + i);   // global_store_b128, NT
    }
    const size_t tail = n4 << 2;
    const size_t tid = blockIdx.x * (size_t)blockDim.x + threadIdx.x;
    if (tail + tid < n) out[tail + tid] = 0.f;
}

// ---------------- Phase 1: gating logits via V_WMMA_F32_16X16X4_F32 ---------
// One wave per 16-token tile. A = x[16 x 4] chunk, B = w_g^T[4 x 16]
// (experts 8..15 are zero padding). K-loop over C in steps of 4.
__global__ __launch_bounds__(32) void router_gating_wmma_kernel(
    const float* __restrict__ x,    // [num_tokens][C]
    const float* __restrict__ wg,   // [8][C]
    float* __restrict__ logits,     // [num_tokens][8]
    int C) {
    const int tile = blockIdx.x;
    const int lane = threadIdx.x;       // 0..31
    const int n    = lane & 15;         // column (expert id, padded to 16)
    const int kh   = lane >> 4;         // K half select (0 or 1)

    // A: lane holds row M = (lane&15); VGPR0/1 = K = 2*kh, 2*kh+1
    const float* arow = x + (size_t)(tile * 16 + n) * (size_t)C;
    // B: lane holds column N = (lane&15); same K striping. Mask padded cols.
    const float  bmsk = (n < N_EXP) ? 1.0f : 0.0f;
    const float* brow = wg + (size_t)(n & (N_EXP - 1)) * (size_t)C;

    v8f acc = {};
    for (int k = 0; k < C; k += 4) {
        v2f a = *(const v2f*)(arow + k + 2 * kh);
        v2f bv = *(const v2f*)(brow + k + 2 * kh);
        v2f b;
        b.x = bv.x * bmsk;
        b.y = bv.y * bmsk;
        // 8 args: (neg_a, A, neg_b, B, c_mod, C, reuse_a, reuse_b)
        acc = __builtin_amdgcn_wmma_f32_16x16x4_f32(
            false, a, false, b, (short)0, acc, false, false);
    }

    // C/D layout: VGPR i, lanes 0-15 -> M=i ; lanes 16-31 -> M=8+i ; N=lane&15
    if (n < N_EXP) {
        const int mbase = tile * 16 + (kh ? 8 : 0);
#pragma unroll
        for (int i = 0; i < 8; ++i)
            logits[(size_t)(mbase + i) * N_EXP + n] = acc[i];
    }
}

// ---------------- Phase 2: top-2 + masked softmax (softmax over 2 values) ---
__global__ void router_topk_softmax_kernel(
    const float* __restrict__ logits,   // [num_tokens][8]
    int* __restrict__ topk_idx,         // [num_tokens][2]
    float* __restrict__ topk_prob,      // [num_tokens][2]
    int num_tokens) {
    const int t = blockIdx.x * blockDim.x + threadIdx.x;
    if (t >= num_tokens) return;
    float l[N_EXP];
#pragma unroll
    for (int e = 0; e < N_EXP; ++e) l[e] = logits[(size_t)t * N_EXP + e];

    // argmax (ties -> lowest index, matching lax.top_k)
    int i1 = 0; float m1 = l[0];
#pragma unroll
    for (int e = 1; e < N_EXP; ++e)
        if (l[e] > m1) { m1 = l[e]; i1 = e; }
    int i2 = (i1 == 0) ? 1 : 0; float m2 = l[i2];
#pragma unroll
    for (int e = 0; e < N_EXP; ++e)
        if (e != i1 && l[e] > m2) { m2 = l[e]; i2 = e; }

    // softmax over {m1, m2} with all else -inf
    const float e2 = expf(m2 - m1);
    const float inv = 1.0f / (1.0f + e2);
    topk_idx[t * 2 + 0] = i1;
    topk_idx[t * 2 + 1] = i2;
    topk_prob[t * 2 + 0] = inv;
    topk_prob[t * 2 + 1] = e2 * inv;
}

// ---------------- Phase 3: per-expert capacity ranking ----------------------
// Reference order is k-major then token: flat index a = k*num_tokens + t.
// One block per expert scans all 2*num_tokens assignments.
__global__ void router_rank_scan_kernel(
    const int* __restrict__ topk_idx,   // [num_tokens][2]
    int* __restrict__ rank,             // [2*num_tokens]
    float* __restrict__ used_cap,       // d_out[0..7]
    int num_tokens, int cap) {
    const int e = blockIdx.x;
    const int tid = threadIdx.x;
    const int total = num_tokens * TOP_K;

    __shared__ int chunk[256];
    __shared__ int base;
    if (tid == 0) base = 0;
    __syncthreads();

    for (int start = 0; start < total; start += 256) {
        const int a = start + tid;
        int match = 0;
        if (a < total) {
            const int k = a / num_tokens;
            const int t = a - k * num_tokens;
            match = (topk_idx[t * 2 + k] == e) ? 1 : 0;
        }
        chunk[tid] = match;
        __syncthreads();
        // Hillis-Steele inclusive scan over 256
        for (int off = 1; off < 256; off <<= 1) {
            int v = (tid >= off) ? chunk[tid - off] : 0;
            __syncthreads();
            chunk[tid] += v;
            __syncthreads();
        }
        const int excl = chunk[tid] - match;
        const int b = base;                    // read before update
        if (match) rank[a] = b + excl;
        __syncthreads();
        if (tid == 0) base += chunk[255];
        __syncthreads();
    }
    if (tid == 0) {
        const int tot = base;
        used_cap[e] = (float)(tot < cap ? tot : cap);
    }
}

// ---------------- Phase 4: scatter weights + mask into zeroed output --------
__global__ void router_scatter_kernel(
    const int* __restrict__ topk_idx, const float* __restrict__ topk_prob,
    const int* __restrict__ rank,
    float* __restrict__ cbw, float* __restrict__ mask,
    int num_tokens, int cap) {
    const int a = blockIdx.x * blockDim.x + threadIdx.x;
    const int total = num_tokens * TOP_K;
    if (a >= total) return;
    const int k = a / num_tokens;
    const int t = a - k * num_tokens;
    const int e = topk_idx[t * 2 + k];
    const int r = rank[a];
    if (r < cap) {
        const float w = topk_prob[t * 2 + k];
        const size_t off = ((size_t)t * N_EXP + e) * (size_t)cap + r;
        cbw[off] = w;
        mask[off] = (w != 0.0f) ? 1.0f : 0.0f;   // sec_mask = cb_weight != 0
    }
}

extern "C" void kernel_launch(void* const* d_in, const int* in_sizes, int n_in,
                              void* d_out, int out_size, void* d_ws, size_t ws_size,
                              hipStream_t stream) {
    const float* x  = (const float*)d_in[0];   // [B*T, C] flat
    const float* wg = (const float*)d_in[1];   // [8, C]

    const int C = in_sizes[1] / N_EXP;                 // 1024
    const int num_tokens = in_sizes[0] / C;            // 4096
    int cap = (int)floorf((float)TOP_K * 2.0f * (float)num_tokens / (float)N_EXP);
    cap += cap & 1;
    if (cap < 4) cap = 4;                              // 2048

    // workspace partition
    char* ws = (char*)d_ws;
    float* logits   = (float*)ws;  ws += (size_t)num_tokens * N_EXP * sizeof(float);
    int*   topk_idx = (int*)ws;    ws += (size_t)num_tokens * 2 * sizeof(int);
    float* topk_p   = (float*)ws;  ws += (size_t)num_tokens * 2 * sizeof(float);
    int*   rank     = (int*)ws;

    float* out_used = (float*)d_out;
    float* out_cbw  = out_used + N_EXP;
    float* out_mask = out_cbw + (size_t)num_tokens * N_EXP * (size_t)cap;

    // 0) zero the entire output (streaming B128 stores; this is the BW roof)
    router_zero_kernel<<<4096, 256, 0, stream>>>((float*)d_out, (size_t)out_size);

    // 1) gating logits on the WMMA pipe (one wave per 16-token tile)
    router_gating_wmma_kernel<<<num_tokens / 16, 32, 0, stream>>>(x, wg, logits, C);

    // 2) top-2 + softmax
    router_topk_softmax_kernel<<<(num_tokens + 255) / 256, 256, 0, stream>>>(
        logits, topk_idx, topk_p, num_tokens);

    // 3) capacity ranking (reference cumsum order) + used_capacity
    router_rank_scan_kernel<<<N_EXP, 256, 0, stream>>>(
        topk_idx, rank, out_used, num_tokens, cap);

    // 4) scatter combine-weights and mask
    router_scatter_kernel<<<(num_tokens * TOP_K + 255) / 256, 256, 0, stream>>>(
        topk_idx, topk_p, rank, out_cbw, out_mask, num_tokens, cap);
}